// GCN_68564857913349
// MI455X (gfx1250) — compile-verified
//
#include <hip/hip_runtime.h>

typedef __attribute__((ext_vector_type(2))) float v2f;
typedef __attribute__((ext_vector_type(8))) float v8f;

#define GCN_N_NODES   100000
#define GCN_N_EDGES   1600000
#define GCN_IN_FEATS  128
#define GCN_HIDDEN    64
#define GCN_CLASSES   40

// ---------------------------------------------------------------------------
// 1) Degree accumulation: deg_src[src[e]] += 1, deg_dst[dst[e]] += 1
// ---------------------------------------------------------------------------
__global__ void gcn_deg_kernel(const int* __restrict__ src,
                               const int* __restrict__ dst,
                               float* __restrict__ deg_src,
                               float* __restrict__ deg_dst, int E) {
    int e = blockIdx.x * blockDim.x + threadIdx.x;
    if (e < E) {
        atomicAdd(&deg_src[src[e]], 1.0f);
        atomicAdd(&deg_dst[dst[e]], 1.0f);
    }
}

// ---------------------------------------------------------------------------
// 2) In-place norm = rsqrt(max(deg, 1))   (covers both deg arrays at once)
// ---------------------------------------------------------------------------
__global__ void gcn_norm_kernel(float* __restrict__ d, int n) {
    int i = blockIdx.x * blockDim.x + threadIdx.x;
    if (i < n) {
        float v = d[i];
        v = v < 1.0f ? 1.0f : v;
        d[i] = rsqrtf(v);
    }
}

// ---------------------------------------------------------------------------
// 3) Row-scaled GEMM via V_WMMA_F32_16X16X4_F32:
//      Y[N,64] = (X * rowscale[:,None]) @ W[Kin,64]
//    Block = 128 threads (4 wave32s). Block computes a 16x64 tile:
//    wave w owns output columns [16w, 16w+16). W staged in LDS.
//    A-frag: lane l -> row m0 + l%16, K = k + 2*(l/16) + j  (j = vgpr idx)
//    B-frag: lane l -> col n0 + l%16, same K mapping
//    C/D  : element r -> row m0 + r + 8*(l/16), col n0 + l%16
// ---------------------------------------------------------------------------
__global__ __launch_bounds__(128) void gcn_gemm_rowscale_wmma(
    const float* __restrict__ X, const float* __restrict__ rowscale,
    const float* __restrict__ W, float* __restrict__ Y, int Kin) {
    __shared__ float sW[GCN_IN_FEATS * GCN_HIDDEN]; // up to 128x64 = 32 KB

    const int tid  = threadIdx.x;
    const int wave = tid >> 5;      // 0..3 -> output col tile
    const int lane = tid & 31;
    const int m0   = blockIdx.x * 16;
    const int n0   = wave * 16;

    for (int idx = tid; idx < Kin * GCN_HIDDEN; idx += 128) sW[idx] = W[idx];
    __syncthreads();

    const int g  = lane >> 4;       // lane half (K offset group)
    const int lm = lane & 15;
    const int m  = m0 + lm;
    const float ns = rowscale[m];
    const float* xrow = X + (size_t)m * Kin;

    v8f acc = {};
    for (int k = 0; k < Kin; k += 4) {
        v2f a = *reinterpret_cast<const v2f*>(xrow + k + 2 * g);
        a.x *= ns;
        a.y *= ns;
        v2f b;
        b.x = sW[(k + 2 * g)     * GCN_HIDDEN + n0 + lm];
        b.y = sW[(k + 2 * g + 1) * GCN_HIDDEN + n0 + lm];
        acc = __builtin_amdgcn_wmma_f32_16x16x4_f32(
            /*neg_a=*/false, a, /*neg_b=*/false, b,
            /*c_mod=*/(short)0, acc, /*reuse_a=*/false, /*reuse_b=*/false);
    }

    float* yt = Y + (size_t)m0 * GCN_HIDDEN + n0 + lm;
#pragma unroll
    for (int r = 0; r < 8; ++r) {
        yt[(size_t)(r + 8 * g) * GCN_HIDDEN] = acc[r];
    }
}

// ---------------------------------------------------------------------------
// 4) Edge scatter-add: h[dst[e], f] += tmp[src[e], f], 64 feats/edge.
//    64 consecutive threads handle one edge -> fully coalesced gather & add.
//    Source and destination are 25.6 MB each -> L2-resident on MI455X.
// ---------------------------------------------------------------------------
__global__ void gcn_scatter_kernel(const float* __restrict__ tmp,
                                   const int* __restrict__ src,
                                   const int* __restrict__ dst,
                                   float* __restrict__ h) {
    unsigned long long idx =
        (unsigned long long)blockIdx.x * blockDim.x + threadIdx.x;
    int e = (int)(idx >> 6);
    int f = (int)(idx & 63);
    if (e < GCN_N_EDGES) {
        float v = tmp[(size_t)src[e] * GCN_HIDDEN + f];
        atomicAdd(&h[(size_t)dst[e] * GCN_HIDDEN + f], v);
    }
}

// ---------------------------------------------------------------------------
// 5) Epilogue: h = relu(h * norm_dst[row] + bias[feat])
// ---------------------------------------------------------------------------
__global__ void gcn_epilogue_kernel(float* __restrict__ h,
                                    const float* __restrict__ ndst,
                                    const float* __restrict__ bias) {
    unsigned long long idx =
        (unsigned long long)blockIdx.x * blockDim.x + threadIdx.x;
    int i = (int)(idx >> 6);
    int f = (int)(idx & 63);
    float v = h[idx] * ndst[i] + bias[f];
    h[idx] = v > 0.0f ? v : 0.0f;
}

// ---------------------------------------------------------------------------
// 6) Final: out[N,40] = [h1|h2|h3] @ Wl[192,40] + bl.
//    96 threads (3 wave32s); wave w owns cols [16w,16w+16) (40 padded to 48).
//    B loads are guarded with a branchless select so EXEC is all-ones at WMMA.
// ---------------------------------------------------------------------------
__global__ __launch_bounds__(96) void gcn_final_wmma(
    const float* __restrict__ h1, const float* __restrict__ h2,
    const float* __restrict__ h3, const float* __restrict__ Wl,
    const float* __restrict__ bl, float* __restrict__ out) {
    const int tid  = threadIdx.x;
    const int wave = tid >> 5;     // 0..2
    const int lane = tid & 31;
    const int m0   = blockIdx.x * 16;
    const int n0   = wave * 16;
    const int g    = lane >> 4;
    const int lm   = lane & 15;
    const int m    = m0 + lm;
    const int n    = n0 + lm;
    const bool nvalid = (n < GCN_CLASSES);

    const float* segs[3] = {h1, h2, h3};

    v8f acc = {};
    for (int k = 0; k < 3 * GCN_HIDDEN; k += 4) {
        const float* seg = segs[k >> 6];
        const int ko = (k & 63) + 2 * g;                 // stays inside segment
        v2f a = *reinterpret_cast<const v2f*>(seg + (size_t)m * GCN_HIDDEN + ko);
        const int kr = k + 2 * g;
        v2f b;
        b.x = nvalid ? Wl[(size_t)kr * GCN_CLASSES + n]       : 0.0f;
        b.y = nvalid ? Wl[(size_t)(kr + 1) * GCN_CLASSES + n] : 0.0f;
        acc = __builtin_amdgcn_wmma_f32_16x16x4_f32(
            false, a, false, b, (short)0, acc, false, false);
    }

    if (nvalid) {
        float bv = bl[n];
#pragma unroll
        for (int r = 0; r < 8; ++r) {
            out[(size_t)(m0 + r + 8 * g) * GCN_CLASSES + n] = acc[r] + bv;
        }
    }
}

// ---------------------------------------------------------------------------
// Launch
// ---------------------------------------------------------------------------
extern "C" void kernel_launch(void* const* d_in, const int* in_sizes, int n_in,
                              void* d_out, int out_size, void* d_ws, size_t ws_size,
                              hipStream_t stream) {
    const float* features = (const float*)d_in[0];
    const float* W1 = (const float*)d_in[1];
    const float* b1 = (const float*)d_in[2];
    const float* W2 = (const float*)d_in[3];
    const float* b2 = (const float*)d_in[4];
    const float* W3 = (const float*)d_in[5];
    const float* b3 = (const float*)d_in[6];
    const float* Wl = (const float*)d_in[7];
    const float* bl = (const float*)d_in[8];
    const int*  src = (const int*)d_in[9];
    const int*  dst = (const int*)d_in[10];
    float* out = (float*)d_out;

    // Workspace layout (floats): [nsrc N][ndst N][tmp 64N][h1 64N][h2 64N][h3 64N]
    float* ws   = (float*)d_ws;
    float* nsrc = ws;
    float* ndst = ws + GCN_N_NODES;
    float* tmp  = ws + 2 * GCN_N_NODES;
    float* h1   = tmp + (size_t)64 * GCN_N_NODES;
    float* h2   = h1  + (size_t)64 * GCN_N_NODES;
    float* h3   = h2  + (size_t)64 * GCN_N_NODES;

    const size_t hbytes = (size_t)64 * GCN_N_NODES * sizeof(float);

    // Degrees -> norms
    hipMemsetAsync(nsrc, 0, (size_t)2 * GCN_N_NODES * sizeof(float), stream);
    gcn_deg_kernel<<<(GCN_N_EDGES + 255) / 256, 256, 0, stream>>>(
        src, dst, nsrc, ndst, GCN_N_EDGES);
    gcn_norm_kernel<<<(2 * GCN_N_NODES + 255) / 256, 256, 0, stream>>>(
        nsrc, 2 * GCN_N_NODES);

    const float* Xs[3] = {features, h1, h2};
    const float* Ws[3] = {W1, W2, W3};
    const float* bs[3] = {b1, b2, b3};
    float*       hs[3] = {h1, h2, h3};
    const int    Ks[3] = {GCN_IN_FEATS, GCN_HIDDEN, GCN_HIDDEN};

    const unsigned scatter_blocks = (unsigned)(((size_t)GCN_N_EDGES * 64 + 255) / 256);
    const unsigned epi_blocks     = (unsigned)(((size_t)GCN_N_NODES * 64 + 255) / 256);

    for (int l = 0; l < 3; ++l) {
        gcn_gemm_rowscale_wmma<<<GCN_N_NODES / 16, 128, 0, stream>>>(
            Xs[l], nsrc, Ws[l], tmp, Ks[l]);
        hipMemsetAsync(hs[l], 0, hbytes, stream);
        gcn_scatter_kernel<<<scatter_blocks, 256, 0, stream>>>(tmp, src, dst, hs[l]);
        gcn_epilogue_kernel<<<epi_blocks, 256, 0, stream>>>(hs[l], ndst, bs[l]);
    }

    gcn_final_wmma<<<GCN_N_NODES / 16, 96, 0, stream>>>(h1, h2, h3, Wl, bl, out);
}